// GCN_66864050864373
// MI455X (gfx1250) — compile-verified
//
#include <hip/hip_runtime.h>
#include <hip/hip_bf16.h>

typedef float v2f __attribute__((ext_vector_type(2)));
typedef float v8f __attribute__((ext_vector_type(8)));

#define IN_C 128

// ---------------- degree / normalization ----------------

__global__ void deg_init(float* deg, int n) {
    int i = blockIdx.x * blockDim.x + threadIdx.x;
    if (i < n) deg[i] = 1.0f;   // self-loop contributes 1
}

__global__ void deg_accum(const int* __restrict__ dst, float* deg, int e) {
    int i = blockIdx.x * blockDim.x + threadIdx.x;
    if (i < e) atomicAdd(&deg[dst[i]], 1.0f);
}

__global__ void deg_rsqrt(float* deg, int n) {
    int i = blockIdx.x * blockDim.x + threadIdx.x;
    if (i < n) deg[i] = rsqrtf(deg[i]);   // deg -> dis = D^{-1/2}
}

// ---------------- WMMA GEMM: C[n x 128] = A[n x 128] * W[128 x 128] ----------------
// Block = 128 threads = 4 waves; each wave owns one 16-row tile (full tiles only).
// W staged once per block into LDS, pair-major so a B fragment is one ds_load_b64:
//   wl[(k>>1)*256 + n*2 + (k&1)] = W[k][n]
// A fragments (16x128 row block) hoisted into 32 x v2f registers per lane.

__global__ void gemm_wmma_f32(const float* __restrict__ A,
                              const float* __restrict__ W,
                              float* __restrict__ C, int full_rows) {
    __shared__ float wl[IN_C * IN_C];            // 64 KB

    // --- stage + swizzle W into LDS (coalesced global reads) ---
    for (int idx = threadIdx.x; idx < IN_C * IN_C; idx += blockDim.x) {
        int k = idx >> 7;           // row of W (K dim)
        int n = idx & 127;          // col of W (N dim)
        wl[(k >> 1) * 256 + n * 2 + (k & 1)] = W[idx];
    }
    __syncthreads();

    int wave = (int)((blockIdx.x * blockDim.x + threadIdx.x) >> 5);
    int lane = threadIdx.x & 31;
    int row0 = wave << 4;
    if (row0 >= full_rows) return;

    int m  = lane & 15;   // A: row within tile; B/C: column within tile
    int kh = lane >> 4;   // lane-half: K pair select (A/B), M half (C/D)

    // --- hoist all 32 A fragments: lane holds A[row0+m][k0+2*kh ..+1] ---
    const float* arow = A + (size_t)(row0 + m) * IN_C;
    v2f afrag[32];
    #pragma unroll
    for (int kk = 0; kk < 32; ++kk)
        afrag[kk] = *(const v2f*)(arow + kk * 4 + (kh << 1));

    #pragma unroll
    for (int nt = 0; nt < 8; ++nt) {
        int n0 = nt * 16;
        v8f acc = {0.f, 0.f, 0.f, 0.f, 0.f, 0.f, 0.f, 0.f};
        #pragma unroll
        for (int kk = 0; kk < 32; ++kk) {
            // B fragment: (W[ka][n0+m], W[ka+1][n0+m]), ka = 4*kk + 2*kh
            int kp = (kk << 1) + kh;                    // k-pair index = ka>>1
            v2f b = *(const v2f*)&wl[kp * 256 + (n0 + m) * 2];
            acc = __builtin_amdgcn_wmma_f32_16x16x4_f32(
                false, afrag[kk], false, b, (short)0, acc, false, false);
        }
        // C/D layout: VGPR r -> row = row0 + 8*kh + r, col = n0 + m
        float* cp = C + (size_t)(row0 + (kh << 3)) * IN_C + n0 + m;
        #pragma unroll
        for (int r = 0; r < 8; ++r) cp[(size_t)r * IN_C] = acc[r];
    }
}

// scalar fallback for tail rows (launched only when n_rows % 16 != 0)
__global__ void gemm_tail(const float* __restrict__ A, const float* __restrict__ W,
                          float* __restrict__ C, int row_start, int n_rows) {
    int idx = blockIdx.x * blockDim.x + threadIdx.x;
    int r = row_start + (idx >> 7);
    int c = idx & 127;
    if (r >= n_rows) return;
    float s = 0.f;
    for (int k = 0; k < IN_C; ++k) s += A[(size_t)r * IN_C + k] * W[(size_t)k * IN_C + c];
    C[(size_t)r * IN_C + c] = s;
}

// ---------------- aggregation ----------------

// out[i][c] = bias[c] + xw[i][c] * dis[i]^2   (self-loop + bias, full overwrite)
__global__ void agg_init(const float* __restrict__ xw, const float* __restrict__ dis,
                         const float* __restrict__ bias, float* __restrict__ out, int n) {
    int idx = blockIdx.x * blockDim.x + threadIdx.x;
    if (idx >= n * IN_C) return;
    int node = idx >> 7;
    int c    = idx & 127;
    float d = dis[node];
    out[idx] = bias[c] + xw[idx] * d * d;
}

// one wave per edge; each lane handles 4 channels (float4 gather + 4 f32 atomics)
__global__ void edge_scatter(const float* __restrict__ xw, const float* __restrict__ dis,
                             const int* __restrict__ src, const int* __restrict__ dst,
                             float* __restrict__ out, int e_count) {
    int wave = (int)((blockIdx.x * blockDim.x + threadIdx.x) >> 5);
    int lane = threadIdx.x & 31;
    if (wave >= e_count) return;
    int s = src[wave];
    int d = dst[wave];
    float nrm = dis[s] * dis[d];
    const float4* xp = (const float4*)(xw + (size_t)s * IN_C);
    float4 v = xp[lane];
    float* op = out + (size_t)d * IN_C + lane * 4;
    atomicAdd(op + 0, v.x * nrm);
    atomicAdd(op + 1, v.y * nrm);
    atomicAdd(op + 2, v.z * nrm);
    atomicAdd(op + 3, v.w * nrm);
}

__global__ void relu_inplace(float* h, int count) {
    int i = blockIdx.x * blockDim.x + threadIdx.x;
    if (i < count) h[i] = fmaxf(h[i], 0.0f);
}

// ---------------- orchestration ----------------

static inline void run_gemm(const float* A, const float* W, float* C, int N,
                            hipStream_t stream) {
    int full_tiles = N / 16;
    int full_rows  = full_tiles * 16;
    if (full_tiles > 0) {
        int blocks = (full_tiles + 3) / 4;   // 4 waves (tiles) per 128-thread block
        gemm_wmma_f32<<<blocks, 128, 0, stream>>>(A, W, C, full_rows);
    }
    int tail = N - full_rows;
    if (tail > 0) {
        int tcount = tail * IN_C;
        gemm_tail<<<(tcount + 255) / 256, 256, 0, stream>>>(A, W, C, full_rows, N);
    }
}

extern "C" void kernel_launch(void* const* d_in, const int* in_sizes, int n_in,
                              void* d_out, int out_size, void* d_ws, size_t ws_size,
                              hipStream_t stream) {
    const float* x  = (const float*)d_in[0];
    const int*   ei = (const int*)  d_in[1];
    const float* W1 = (const float*)d_in[2];
    const float* b1 = (const float*)d_in[3];
    const float* W2 = (const float*)d_in[4];
    const float* b2 = (const float*)d_in[5];
    const float* W3 = (const float*)d_in[6];
    const float* b3 = (const float*)d_in[7];

    const int N = in_sizes[0] / IN_C;     // 100000
    const int E = in_sizes[1] / 2;        // 1600000
    const int* src = ei;
    const int* dst = ei + E;

    float* out = (float*)d_out;           // N x 128, reused as inter-layer buffer

    // workspace layout: dis[N] (padded) | buf0[N*128]
    float* ws_f = (float*)d_ws;
    size_t dis_elems = ((size_t)N + 255) & ~(size_t)255;
    float* dis  = ws_f;
    float* buf0 = ws_f + dis_elems;
    (void)ws_size;

    const int elem = N * IN_C;
    const int scat_blocks = (int)(((size_t)E * 32 + 255) / 256);
    const int elem_blocks = (elem + 255) / 256;

    // --- normalization: dis = rsqrt(1 + indegree) ---
    deg_init <<<(N + 255) / 256, 256, 0, stream>>>(dis, N);
    deg_accum<<<(E + 255) / 256, 256, 0, stream>>>(dst, dis, E);
    deg_rsqrt<<<(N + 255) / 256, 256, 0, stream>>>(dis, N);

    // --- layer 1: x -> out ---
    run_gemm(x, W1, buf0, N, stream);
    agg_init     <<<elem_blocks, 256, 0, stream>>>(buf0, dis, b1, out, N);
    edge_scatter <<<scat_blocks, 256, 0, stream>>>(buf0, dis, src, dst, out, E);
    relu_inplace <<<elem_blocks, 256, 0, stream>>>(out, elem);

    // --- layer 2: out -> out ---
    run_gemm(out, W2, buf0, N, stream);
    agg_init     <<<elem_blocks, 256, 0, stream>>>(buf0, dis, b2, out, N);
    edge_scatter <<<scat_blocks, 256, 0, stream>>>(buf0, dis, src, dst, out, E);
    relu_inplace <<<elem_blocks, 256, 0, stream>>>(out, elem);

    // --- layer 3: out -> out (no relu) ---
    run_gemm(out, W3, buf0, N, stream);
    agg_init     <<<elem_blocks, 256, 0, stream>>>(buf0, dis, b3, out, N);
    edge_scatter <<<scat_blocks, 256, 0, stream>>>(buf0, dis, src, dst, out, E);
}